// TabICL_85907935855183
// MI455X (gfx1250) — compile-verified
//
#include <hip/hip_runtime.h>
#include <hip/hip_bf16.h>
#include <math.h>

typedef __attribute__((ext_vector_type(2))) float v2f;
typedef __attribute__((ext_vector_type(8))) float v8f;

// Problem constants (match reference setup_inputs)
constexpr int BATCH   = 4;
constexpr int TROWS   = 4608;   // T
constexpr int DIM     = 512;    // D
constexpr int NTRAIN  = 4096;
constexpr int NTEST   = 512;    // T - train_size
constexpr int KNN     = 32;     // k
constexpr int CHUNK   = 512;    // train rows processed per LDS chunk
constexpr int NCHUNK  = NTRAIN / CHUNK;          // 8
constexpr int UW      = CHUNK + KNN;             // 544 = union width (chunk + running best)
constexpr int DSTRIDE = UW + 4;                  // 548, pad for LDS banks

// d_out layout (flat floats, reference tuple order)
constexpr size_t OFF_ICL = 0;
constexpr size_t OFF_Y   = (size_t)BATCH * NTEST * (KNN + 1) * DIM;  // 34,603,008
constexpr size_t OFF_D   = OFF_Y + (size_t)BATCH * NTEST * KNN;      // 34,668,544

// ---------------------------------------------------------------------------
// Kernel 1: squared L2 norm of every representation row (train+test), 1 wave/row
// ---------------------------------------------------------------------------
__global__ __launch_bounds__(256)
void row_norms_kernel(const float* __restrict__ reps, float* __restrict__ norms)
{
    const int row  = blockIdx.x * 8 + (threadIdx.x >> 5);
    const int lane = threadIdx.x & 31;
    const float* p = reps + (size_t)row * DIM;
    float s = 0.f;
#pragma unroll
    for (int i = 0; i < DIM / 32; ++i) {
        float v = p[lane + 32 * i];
        s = fmaf(v, v, s);
    }
#pragma unroll
    for (int off = 16; off > 0; off >>= 1)
        s += __shfl_xor(s, off, 32);
    if (lane == 0) norms[row] = s;
}

// ---------------------------------------------------------------------------
// Kernel 2 (fused): WMMA cross-term GEMM -> dists in LDS -> streaming top-32
// merge per row -> write knn_dists / labels -> gather icl_input.
// One workgroup = 16 waves = one (batch, 16-test-row) tile. Grid = 128.
// ---------------------------------------------------------------------------

// Load one K=16 slab of fragments (4 sub-steps of K=4): A shared, B for 2 tiles.
#define LOAD_SLAB(A, B0, B1, KOFF)                                   \
    _Pragma("unroll")                                                \
    for (int u = 0; u < 4; ++u) {                                    \
        A[u]  = *(const v2f*)(aP  + (KOFF) + 4 * u);                 \
        B0[u] = *(const v2f*)(bP0 + (KOFF) + 4 * u);                 \
        B1[u] = *(const v2f*)(bP1 + (KOFF) + 4 * u);                 \
    }

// 4 K=4 WMMAs into two independent accumulator chains (interleaved).
#define WMMA_SLAB(A, B0, B1)                                                   \
    _Pragma("unroll")                                                          \
    for (int u = 0; u < 4; ++u) {                                              \
        acc0 = __builtin_amdgcn_wmma_f32_16x16x4_f32(                          \
                   false, A[u], false, B0[u], (short)0, acc0, false, false);   \
        acc1 = __builtin_amdgcn_wmma_f32_16x16x4_f32(                          \
                   false, A[u], false, B1[u], (short)0, acc1, false, false);   \
    }

__global__ __launch_bounds__(512)
void tabicl_fused_kernel(const float* __restrict__ reps,
                         const int*   __restrict__ ytrain,
                         const float* __restrict__ norms,
                         float*       __restrict__ out)
{
    __shared__ float distBuf[16][DSTRIDE];  // [row][0..511]=chunk dists, [512..543]=running best
    __shared__ int   idxArr[16][KNN];       // global train indices of running best

    const int wg   = blockIdx.x;            // 0..127
    const int b    = wg >> 5;               // batch
    const int t0   = (wg & 31) * 16;        // first test row of tile
    const int tid  = threadIdx.x;
    const int wave = tid >> 5;              // 0..15
    const int lane = tid & 31;
    const int hm   = lane & 15;             // M (A) / N (B) index within tile
    const int hh   = lane >> 4;             // lane-half selects K offset 2

    const float* bRep     = reps + (size_t)b * TROWS * DIM;
    const float* trainRep = bRep;                                // rows [0,4096)
    const float* testRep  = bRep + (size_t)NTRAIN * DIM;         // rows [4096,4608)
    const float* rn       = norms + (size_t)b * TROWS;           // train norms
    const float* tn       = rn + NTRAIN;                         // test norms

    // init running-best slots to +inf
    if (tid < 16 * KNN)
        distBuf[tid >> 5][CHUNK + (tid & 31)] = __builtin_inff();
    __syncthreads();

    // preload this lane's 8 test norms: output element j maps to m = j + 8*hh
    float tnv[8];
#pragma unroll
    for (int j = 0; j < 8; ++j) tnv[j] = tn[t0 + j + 8 * hh];

    float bd = __builtin_inff();   // this lane's held winner (dist) for row 'wave'
    int   bi = 0;                  // and its global train index

    const float* aP = testRep + (size_t)(t0 + hm) * DIM + 2 * hh;

    for (int c = 0; c < NCHUNK; ++c) {
        const int sBase = c * CHUNK;

        // ---- GEMM: wave computes a 16x32 strip (two 16x16 tiles, shared A) ----
        {
            const int s0 = sBase + wave * 32;       // first tile col base
            const float* bP0 = trainRep + (size_t)(s0 + hm) * DIM + 2 * hh;
            const float* bP1 = bP0 + 16 * DIM;

            v8f acc0 = {0.f, 0.f, 0.f, 0.f, 0.f, 0.f, 0.f, 0.f};
            v8f acc1 = {0.f, 0.f, 0.f, 0.f, 0.f, 0.f, 0.f, 0.f};

            v2f aA[4], bA[4], cA[4];   // ping buffer
            v2f aB[4], bB[4], cB[4];   // pong buffer

            LOAD_SLAB(aA, bA, cA, 0)
            for (int k0 = 0; k0 < DIM; k0 += 32) {
                LOAD_SLAB(aB, bB, cB, k0 + 16)      // prefetch pong
                WMMA_SLAB(aA, bA, cA)               // compute ping
                if (k0 + 32 < DIM) {
                    LOAD_SLAB(aA, bA, cA, k0 + 32)  // prefetch next ping
                }
                WMMA_SLAB(aB, bB, cB)               // compute pong
            }

            // dist = sqrt(max(tn + rn - 2*dot, 0)); v_sqrt_f32 is plenty here
            const float rnv0 = rn[s0 + hm];
            const float rnv1 = rn[s0 + 16 + hm];
#pragma unroll
            for (int j = 0; j < 8; ++j) {
                const int m = j + 8 * hh;           // test row within tile
                float v0 = tnv[j] + rnv0 - 2.0f * acc0[j];
                float v1 = tnv[j] + rnv1 - 2.0f * acc1[j];
                distBuf[m][wave * 32 + hm]      = __builtin_amdgcn_sqrtf(fmaxf(v0, 0.0f));
                distBuf[m][wave * 32 + 16 + hm] = __builtin_amdgcn_sqrtf(fmaxf(v1, 0.0f));
            }
        }
        __syncthreads();

        // ---- merge: wave w extracts top-32 of (chunk ∪ running best) for row w ----
        {
            const int row = wave;
            float dl[UW / 32];                    // 17 union entries per lane, in regs
#pragma unroll
            for (int u = 0; u < UW / 32; ++u)
                dl[u] = distBuf[row][lane + 32 * u];

            for (int i = 0; i < KNN; ++i) {
                float md = __builtin_inff();
                int   mp = UW;
#pragma unroll
                for (int u = 0; u < UW / 32; ++u) {
                    const int p = lane + 32 * u;
                    if (dl[u] < md || (dl[u] == md && p < mp)) { md = dl[u]; mp = p; }
                }
#pragma unroll
                for (int off = 16; off > 0; off >>= 1) {
                    float od = __shfl_xor(md, off, 32);
                    int   op = __shfl_xor(mp, off, 32);
                    if (od < md || (od == md && op < mp)) { md = od; mp = op; }
                }
                if (lane == (mp & 31)) dl[mp >> 5] = __builtin_inff();  // remove winner
                if (lane == i) {                                        // lane i keeps winner i
                    bd = md;
                    bi = (mp < CHUNK) ? (sBase + mp) : idxArr[row][mp - CHUNK];
                }
            }
            // publish new running best (sorted ascending by construction)
            distBuf[row][CHUNK + lane] = bd;
            idxArr[row][lane]          = bi;
        }
        __syncthreads();
    }

    // ---- emit knn_dists (ascending) and gathered labels ----
    {
        const int t = t0 + wave;
        const size_t btRow = (size_t)(b * NTEST + t);
        out[OFF_D + btRow * KNN + lane] = bd;
        out[OFF_Y + btRow * KNN + lane] = (float)ytrain[b * NTRAIN + bi];
    }
    // idxArr finalized before last __syncthreads inside the loop.

    // ---- gather icl_input: 16 rows x 33 segments x 512 floats, float4 stores ----
    {
        const int grp = tid >> 7;     // 4 groups of 128 lanes
        const int gl  = tid & 127;    // 128 float4 per 512-float segment
        for (int seg = grp; seg < 16 * (KNN + 1); seg += 4) {
            const int r = seg / (KNN + 1);
            const int j = seg - r * (KNN + 1);
            const float4* src = (j < KNN)
                ? (const float4*)(trainRep + (size_t)idxArr[r][j] * DIM)
                : (const float4*)(testRep  + (size_t)(t0 + r) * DIM);
            float4* dst = (float4*)(out + OFF_ICL +
                          ((size_t)(b * NTEST + t0 + r) * (KNN + 1) + j) * DIM);
            dst[gl] = src[gl];
        }
    }
}

// ---------------------------------------------------------------------------
extern "C" void kernel_launch(void* const* d_in, const int* in_sizes, int n_in,
                              void* d_out, int out_size, void* d_ws, size_t ws_size,
                              hipStream_t stream)
{
    (void)in_sizes; (void)n_in; (void)out_size; (void)ws_size;
    const float* reps = (const float*)d_in[0];   // (4,4608,512) f32
    const int*   ytr  = (const int*)d_in[1];     // (4,4096) i32
    float* norms = (float*)d_ws;                 // 18432 floats (73,728 B)
    float* out   = (float*)d_out;

    row_norms_kernel<<<(BATCH * TROWS) / 8, 256, 0, stream>>>(reps, norms);
    tabicl_fused_kernel<<<BATCH * (NTEST / 16), 512, 0, stream>>>(reps, ytr, norms, out);
}